// DiffusionGraphConv_89936615178296
// MI455X (gfx1250) — compile-verified
//
#include <hip/hip_runtime.h>
#include <hip/hip_bf16.h>

typedef float v2f __attribute__((ext_vector_type(2)));
typedef float v8f __attribute__((ext_vector_type(8)));

// ---------------------------------------------------------------------------
// 1) degree init: deg[i] = 1.0 (self loop)
// ---------------------------------------------------------------------------
__global__ void k_init_deg(float* __restrict__ deg, int n) {
    int i = blockIdx.x * blockDim.x + threadIdx.x;
    if (i < n) deg[i] = 1.0f;
}

// 2) degree accumulate over edges: deg[dst[e]] += 1
__global__ void k_deg_scatter(const long long* __restrict__ dst,
                              float* __restrict__ deg, int e) {
    int i = blockIdx.x * blockDim.x + threadIdx.x;
    if (i < e) atomicAdd(&deg[(int)dst[i]], 1.0f);
}

// 3) dinv[i] = rsqrt(deg[i])  (deg >= 1 always, no zero check needed)
__global__ void k_rsqrt(float* __restrict__ deg, int n) {
    int i = blockIdx.x * blockDim.x + threadIdx.x;
    if (i < n) deg[i] = rsqrtf(deg[i]);
}

// ---------------------------------------------------------------------------
// 4) h = x @ (W1 + W2)    [N x 32] = [N x 32][32 x 32]
//    WMMA f32 16x16x4, K=32 via 8 chained WMMAs. Block = 128 thr = 4 waves:
//    wave -> (rowTile 0/1) x (colTile 0/1), block covers 32 rows.
// ---------------------------------------------------------------------------
__global__ void k_gemm_h(const float* __restrict__ x,
                         const float* __restrict__ W1,
                         const float* __restrict__ W2,
                         float* __restrict__ h, int n) {
    __shared__ float Ws[32 * 33];   // padded: stride 33 floats -> no bank conflicts

    int t = threadIdx.x;
    for (int i = t; i < 32 * 32; i += blockDim.x) {
        int r = i >> 5, c = i & 31;
        Ws[r * 33 + c] = W1[i] + W2[i];
    }
    __syncthreads();

    const int wave = t >> 5;           // 0..3
    const int lane = t & 31;
    const int m    = lane & 15;        // M (A) / N (B,C) position
    const int hi   = lane >> 4;        // half-wave select

    const int row0 = (blockIdx.x * 2 + (wave >> 1)) * 16;  // 16-row tile base
    const int col0 = (wave & 1) * 16;                      // 16-col tile base
    if (row0 + 16 > n) return;         // wave-uniform: EXEC stays all-ones

    const float* xrow = x + (size_t)(row0 + m) * 32;
    const int    col  = col0 + m;

    v8f acc = {0.f, 0.f, 0.f, 0.f, 0.f, 0.f, 0.f, 0.f};
    #pragma unroll
    for (int kk = 0; kk < 8; ++kk) {
        const int k0 = kk * 4 + hi * 2;            // this lane's K pair
        v2f a, b;
        a.x = xrow[k0];                            // A[m][k0]
        a.y = xrow[k0 + 1];                        // A[m][k0+1]
        b.x = Ws[k0 * 33 + col];                   // B[k0][col]
        b.y = Ws[(k0 + 1) * 33 + col];             // B[k0+1][col]
        acc = __builtin_amdgcn_wmma_f32_16x16x4_f32(
            /*neg_a=*/false, a, /*neg_b=*/false, b,
            /*c_mod=*/(short)0, acc, /*reuse_a=*/false, /*reuse_b=*/false);
    }

    // C/D layout: VGPR j -> row (j + 8*hi), col m
    #pragma unroll
    for (int j = 0; j < 8; ++j) {
        h[(size_t)(row0 + j + 8 * hi) * 32 + col] = acc[j];
    }
}

// ---------------------------------------------------------------------------
// 5) seed out with self-loop term: out[i][:] = dinv[i]^2 * h[i][:]
//    (also deterministically overwrites the poisoned d_out each call)
// ---------------------------------------------------------------------------
__global__ void k_init_out(const float* __restrict__ h,
                           const float* __restrict__ dinv,
                           float* __restrict__ out, int n) {
    int t = blockIdx.x * blockDim.x + threadIdx.x;   // one float4 per thread
    int i = t >> 3;                                  // node
    if (i >= n) return;
    float w = dinv[i] * dinv[i];
    const float4* h4 = (const float4*)h;
    float4 v = h4[t];
    float4* o4 = (float4*)out;
    o4[t] = make_float4(v.x * w, v.y * w, v.z * w, v.w * w);
}

// ---------------------------------------------------------------------------
// 6) edge scatter: out[dst] += dinv[src]*dinv[dst] * h[src]
//    8 threads per edge, each owns a float4 channel chunk. All operands are
//    L2-resident (192 MB L2 >> 52 MB working set) -> L2 atomic units.
// ---------------------------------------------------------------------------
__global__ void k_edge_scatter(const long long* __restrict__ src,
                               const long long* __restrict__ dst,
                               const float* __restrict__ h,
                               const float* __restrict__ dinv,
                               float* __restrict__ out, int e) {
    int t = blockIdx.x * blockDim.x + threadIdx.x;
    int edge = t >> 3;
    if (edge >= e) return;
    int q = t & 7;                       // float4 chunk 0..7
    int s = (int)src[edge];
    int d = (int)dst[edge];
    float w = dinv[s] * dinv[d];
    const float4* h4 = (const float4*)h;
    float4 v = h4[s * 8 + q];
    float* o = out + (size_t)d * 32 + q * 4;
    atomicAdd(o + 0, v.x * w);
    atomicAdd(o + 1, v.y * w);
    atomicAdd(o + 2, v.z * w);
    atomicAdd(o + 3, v.w * w);
}

// ---------------------------------------------------------------------------
// 7) epilogue: out = relu(out + b1 + b2), in place
// ---------------------------------------------------------------------------
__global__ void k_relu_bias(float* __restrict__ out,
                            const float* __restrict__ b1,
                            const float* __restrict__ b2, int total) {
    int i = blockIdx.x * blockDim.x + threadIdx.x;
    if (i >= total) return;
    int c = i & 31;
    float v = out[i] + b1[c] + b2[c];
    out[i] = v > 0.f ? v : 0.f;
}

extern "C" void kernel_launch(void* const* d_in, const int* in_sizes, int n_in,
                              void* d_out, int out_size, void* d_ws, size_t ws_size,
                              hipStream_t stream) {
    const float*     x   = (const float*)d_in[0];
    const long long* ei  = (const long long*)d_in[1];   // int64 [2, E]
    const float*     W1  = (const float*)d_in[2];
    const float*     b1  = (const float*)d_in[3];
    const float*     W2  = (const float*)d_in[4];
    const float*     b2  = (const float*)d_in[5];
    float*           out = (float*)d_out;

    const int N = in_sizes[0] / 32;
    const int E = in_sizes[1] / 2;
    const long long* src = ei;          // edge_index[0]
    const long long* dst = ei + E;      // edge_index[1]

    // workspace layout: [ dinv : N floats ][ h : N*32 floats ]
    float* dinv = (float*)d_ws;
    float* h    = dinv + N;

    const int B = 256;

    // degree + normalization
    k_init_deg<<<(N + B - 1) / B, B, 0, stream>>>(dinv, N);
    k_deg_scatter<<<(E + B - 1) / B, B, 0, stream>>>(dst, dinv, E);
    k_rsqrt<<<(N + B - 1) / B, B, 0, stream>>>(dinv, N);

    // h = x @ (W1 + W2) via WMMA (32 rows per 128-thread block)
    k_gemm_h<<<(N + 31) / 32, 128, 0, stream>>>(x, W1, W2, h, N);

    // self-loop seed of output
    k_init_out<<<(N * 8 + B - 1) / B, B, 0, stream>>>(h, dinv, out, N);

    // edge scatter-add (8 threads / edge)
    {
        long long tot = (long long)E * 8;
        k_edge_scatter<<<(int)((tot + B - 1) / B), B, 0, stream>>>(src, dst, h, dinv, out, E);
    }

    // bias + relu
    k_relu_bias<<<(N * 32 + B - 1) / B, B, 0, stream>>>(out, b1, b2, N * 32);
}